// SPDnet_89318139887819
// MI455X (gfx1250) — compile-verified
//
#include <hip/hip_runtime.h>
#include <hip/hip_bf16.h>

typedef __attribute__((ext_vector_type(2))) float v2f;
typedef __attribute__((ext_vector_type(8))) float v8f;

#define DIM   96
#define D2    24
#define EPSV  1e-4f

// ---------------------------------------------------------------------------
// Setup kernel (1 block): fold the linear algebra that is batch-independent.
//   Wc   = W1 @ W2                (96x24)
//   Meff = lw3 @ lw2 @ lw1        (2x300)
//   beff = lw3@lw2@lb1 + lw3@lb2 + lb3   (2)
// ---------------------------------------------------------------------------
__global__ void spd_setup_kernel(const float* __restrict__ W1,
                                 const float* __restrict__ W2,
                                 const float* __restrict__ lw1,
                                 const float* __restrict__ lb1,
                                 const float* __restrict__ lw2,
                                 const float* __restrict__ lb2,
                                 const float* __restrict__ lw3,
                                 const float* __restrict__ lb3,
                                 float* __restrict__ Wc,
                                 float* __restrict__ Meff,
                                 float* __restrict__ beff) {
    __shared__ float T1[2 * 64];
    const int tid = threadIdx.x;

    // Wc = W1(96x48) @ W2(48x24)
    for (int idx = tid; idx < 96 * 24; idx += 256) {
        const int i = idx / 24, j = idx % 24;
        float s = 0.0f;
        for (int k = 0; k < 48; ++k) s += W1[i * 48 + k] * W2[k * 24 + j];
        Wc[idx] = s;
    }
    // T1 = lw3(2x32) @ lw2(32x64)
    for (int idx = tid; idx < 2 * 64; idx += 256) {
        const int i = idx / 64, j = idx % 64;
        float s = 0.0f;
        for (int k = 0; k < 32; ++k) s += lw3[i * 32 + k] * lw2[k * 64 + j];
        T1[idx] = s;
    }
    __syncthreads();
    // Meff = T1(2x64) @ lw1(64x300)
    for (int idx = tid; idx < 2 * 300; idx += 256) {
        const int i = idx / 300, j = idx % 300;
        float s = 0.0f;
        for (int k = 0; k < 64; ++k) s += T1[i * 64 + k] * lw1[k * 300 + j];
        Meff[idx] = s;
    }
    if (tid < 2) {
        float s = lb3[tid];
        for (int k = 0; k < 64; ++k) s += T1[tid * 64 + k] * lb1[k];
        for (int k = 0; k < 32; ++k) s += lw3[tid * 32 + k] * lb2[k];
        beff[tid] = s;
    }
}

// ---------------------------------------------------------------------------
// Main kernel: one workgroup (256 threads = 8 waves) per batch element.
//   1) T = X @ Wc   (96x24, WMMA f32 16x16x4, X read straight from HBM once)
//   2) Y = Wc^T @ T (24x24, WMMA f32 16x16x4)
//   3) Jacobi eigensolve of Y in LDS (9 sweeps, tournament ordering)
//   4) feats = upper-tri of V diag(log max(w,EPS)) V^T  (300 values)
//   5) out = Meff @ feats + beff   (2 values)
// ---------------------------------------------------------------------------
__global__ void __launch_bounds__(256)
spd_main_kernel(const float* __restrict__ Xg,
                const float* __restrict__ Wc,
                const float* __restrict__ Meff,
                const float* __restrict__ beff,
                float* __restrict__ out) {
    __shared__ float Wcs[96][32];   // Wc zero-padded to 32 cols
    __shared__ float Ts[96][32];    // T = X @ Wc (padded cols are exact zeros)
    __shared__ float As[24][26];    // working symmetric matrix
    __shared__ float Vs[24][26];    // accumulated eigenvectors (columns)
    __shared__ float cw[12], sw[12];
    __shared__ int   pr[12], qr[12];
    __shared__ float lwv[24];
    __shared__ float feats[304];
    __shared__ float red[64];

    const int tid  = threadIdx.x;
    const int wv   = tid >> 5;
    const int lane = tid & 31;
    const int lr   = lane & 15;     // row/col-within-tile index
    const int kh   = lane >> 4;     // K-half selector (0/1)
    const int bb   = blockIdx.x;
    const float* Xb = Xg + (size_t)bb * (DIM * DIM);

    // Stage Wc into LDS, zero-padded to 32 columns.
    for (int idx = tid; idx < 96 * 32; idx += 256) {
        const int r = idx >> 5, c = idx & 31;
        Wcs[r][c] = (c < 24) ? Wc[r * 24 + c] : 0.0f;
    }
    __syncthreads();

    // ---- GEMM1: T(96x32) = X(96x96) @ Wc(96x32). Waves 0..5 own 16-row strips;
    //      each wave reuses its A fragment for both 16-wide N tiles.
    if (wv < 6) {
        v8f acc0 = {};
        v8f acc1 = {};
        const int m = wv * 16 + lr;
        const float* xrow = Xb + m * DIM;
        for (int kk = 0; kk < 24; ++kk) {
            const int k0 = kk * 4 + kh * 2;
            v2f a;  a.x  = xrow[k0];          a.y  = xrow[k0 + 1];
            v2f b0; b0.x = Wcs[k0][lr];       b0.y = Wcs[k0 + 1][lr];
            v2f b1; b1.x = Wcs[k0][16 + lr];  b1.y = Wcs[k0 + 1][16 + lr];
            acc0 = __builtin_amdgcn_wmma_f32_16x16x4_f32(
                false, a, false, b0, (short)0, acc0, false, false);
            acc1 = __builtin_amdgcn_wmma_f32_16x16x4_f32(
                false, a, false, b1, (short)0, acc1, false, false);
        }
        for (int r = 0; r < 8; ++r) {
            const int M = wv * 16 + r + 8 * kh;
            Ts[M][lr]      = acc0[r];
            Ts[M][16 + lr] = acc1[r];
        }
    }
    __syncthreads();

    // ---- GEMM2: Y(24x24) = Wc^T(24x96) @ T(96x24). Waves 0..3 -> 2x2 tiles.
    if (wv < 4) {
        const int mt = wv >> 1, nt = wv & 1;
        v8f acc = {};
        const int mm = mt * 16 + lr;   // row of Wc^T == column of Wc (zero if >=24)
        const int nn = nt * 16 + lr;
        for (int kk = 0; kk < 24; ++kk) {
            const int k0 = kk * 4 + kh * 2;
            v2f a;  a.x  = Wcs[k0][mm]; a.y  = Wcs[k0 + 1][mm];
            v2f b2; b2.x = Ts[k0][nn];  b2.y = Ts[k0 + 1][nn];
            acc = __builtin_amdgcn_wmma_f32_16x16x4_f32(
                false, a, false, b2, (short)0, acc, false, false);
        }
        for (int r = 0; r < 8; ++r) {
            const int M = mt * 16 + r + 8 * kh;
            if (M < 24 && nn < 24) As[M][nn] = acc[r];
        }
    }
    // V = I
    for (int idx = tid; idx < 24 * 24; idx += 256) {
        const int r = idx / 24, c = idx % 24;
        Vs[r][c] = (r == c) ? 1.0f : 0.0f;
    }
    __syncthreads();

    // ---- Cyclic Jacobi, round-robin tournament: 12 disjoint rotations/round,
    //      23 rounds/sweep, 9 sweeps (quadratic convergence; fp32 by sweep ~7).
    for (int sweep = 0; sweep < 9; ++sweep) {
        for (int round = 0; round < 23; ++round) {
            if (tid < 12) {
                const int k  = tid;
                const int i1 = (k == 0) ? 0 : (1 + ((k - 1 + round) % 23));
                const int i2 = 1 + ((22 - k + round) % 23);
                const int p = (i1 < i2) ? i1 : i2;
                const int q = (i1 < i2) ? i2 : i1;
                pr[tid] = p; qr[tid] = q;
                const float app = As[p][p], aqq = As[q][q], apq = As[p][q];
                float c = 1.0f, s = 0.0f;
                if (fabsf(apq) > 1e-14f) {
                    const float theta = 0.5f * (aqq - app) / apq;
                    const float t = copysignf(1.0f, theta) /
                                    (fabsf(theta) + sqrtf(1.0f + theta * theta));
                    c = 1.0f / sqrtf(1.0f + t * t);
                    s = t * c;
                }
                cw[tid] = c; sw[tid] = s;
            }
            __syncthreads();
            // Row phase: A <- J^T A
            for (int idx = tid; idx < 288; idx += 256) {
                const int j = idx / 24, col = idx % 24;
                const int p = pr[j], q = qr[j];
                const float c = cw[j], s = sw[j];
                const float ap = As[p][col], aq = As[q][col];
                As[p][col] = c * ap - s * aq;
                As[q][col] = s * ap + c * aq;
            }
            __syncthreads();
            // Column phase: A <- A J, V <- V J
            for (int idx = tid; idx < 288; idx += 256) {
                const int j = idx / 24, row = idx % 24;
                const int p = pr[j], q = qr[j];
                const float c = cw[j], s = sw[j];
                const float ap = As[row][p], aq = As[row][q];
                As[row][p] = c * ap - s * aq;
                As[row][q] = s * ap + c * aq;
                const float vp = Vs[row][p], vq = Vs[row][q];
                Vs[row][p] = c * vp - s * vq;
                Vs[row][q] = s * vp + c * vq;
            }
            __syncthreads();
        }
    }

    // ---- log of clamped eigenvalues (exact composition of ReEig + LogEig at 24x24)
    if (tid < 24) lwv[tid] = logf(fmaxf(As[tid][tid], EPSV));
    __syncthreads();

    // ---- Upper-triangular features of V diag(lw) V^T  (np.triu_indices order)
    for (int f = tid; f < 300; f += 256) {
        int i = 0, rem = f;
        while (rem >= 24 - i) { rem -= 24 - i; ++i; }
        const int j = i + rem;
        float s = 0.0f;
        for (int k = 0; k < 24; ++k) s += Vs[i][k] * lwv[k] * Vs[j][k];
        feats[f] = s;
    }
    __syncthreads();

    // ---- Folded MLP: out(2) = Meff(2x300) @ feats + beff
    if (tid < 64) {
        const int j = tid >> 5, l = tid & 31;
        float s = 0.0f;
        for (int f = l; f < 300; f += 32) s += Meff[j * 300 + f] * feats[f];
        red[tid] = s;
    }
    __syncthreads();
    if (tid < 2) {
        float s = beff[tid];
        for (int l = 0; l < 32; ++l) s += red[tid * 32 + l];
        out[bb * 2 + tid] = s;
    }
}

// ---------------------------------------------------------------------------
extern "C" void kernel_launch(void* const* d_in, const int* in_sizes, int n_in,
                              void* d_out, int out_size, void* d_ws, size_t ws_size,
                              hipStream_t stream) {
    const float* x   = (const float*)d_in[0];
    const float* W1  = (const float*)d_in[1];
    const float* W2  = (const float*)d_in[2];
    const float* lw1 = (const float*)d_in[3];
    const float* lb1 = (const float*)d_in[4];
    const float* lw2 = (const float*)d_in[5];
    const float* lb2 = (const float*)d_in[6];
    const float* lw3 = (const float*)d_in[7];
    const float* lb3 = (const float*)d_in[8];
    float* out = (float*)d_out;

    float* ws   = (float*)d_ws;
    float* Wc   = ws;            // 96*24  = 2304
    float* Meff = ws + 2304;     // 2*300  = 600
    float* beff = ws + 2904;     // 2

    const int nB = in_sizes[0] / (DIM * DIM);

    spd_setup_kernel<<<1, 256, 0, stream>>>(W1, W2, lw1, lb1, lw2, lb2,
                                            lw3, lb3, Wc, Meff, beff);
    spd_main_kernel<<<nB, 256, 0, stream>>>(x, Wc, Meff, beff, out);
}